// StaticSparseGAT_8495445311613
// MI455X (gfx1250) — compile-verified
//
#include <hip/hip_runtime.h>
#include <cstdint>

#define D 128
#define NH 8

typedef __attribute__((ext_vector_type(2))) float v2f;
typedef __attribute__((ext_vector_type(8))) float v8f;

__device__ __forceinline__ void atomicMaxF(float* addr, float val) {
    // ordered-int trick: correct for any mix of signs
    if (val >= 0.0f) atomicMax((int*)addr, __float_as_int(val));
    else             atomicMin((unsigned int*)addr, __float_as_uint(val));
}

// ---------------------------------------------------------------------------
// 0) workspace init: agg=0, s=0, m=-inf
// ---------------------------------------------------------------------------
__global__ void init_ws(float* __restrict__ agg, float* __restrict__ sbuf,
                        float* __restrict__ mbuf, int N) {
    size_t idx = (size_t)blockIdx.x * blockDim.x + threadIdx.x;
    size_t total = (size_t)N * D;
    if (idx < total) agg[idx] = 0.0f;
    if (idx < (size_t)N * NH) {
        sbuf[idx] = 0.0f;
        mbuf[idx] = __uint_as_float(0xFF800000u);  // -inf
    }
}

// ---------------------------------------------------------------------------
// 1) fused node transforms: Hw1=H@W1, Hw2=H@W2, V=H@Wv, R=H@Wres+b_res
//    one block = 16 node rows; H tile in LDS; 4 waves, one matrix each;
//    f32 WMMA 16x16x4, K=128 in 32 steps.
// ---------------------------------------------------------------------------
__global__ __launch_bounds__(128) void node_gemm(
    const float* __restrict__ H,
    const float* __restrict__ W1, const float* __restrict__ W2,
    const float* __restrict__ Wv, const float* __restrict__ Wr,
    const float* __restrict__ b_res,
    float* __restrict__ Hw1, float* __restrict__ Hw2,
    float* __restrict__ Vv,  float* __restrict__ R, int N)
{
    __shared__ float As[16 * 132];   // 16 rows of H, stride 132 (bank-conflict free)
    const int t = threadIdx.x;
    const int row0 = blockIdx.x * 16;
    {
        const int r  = t >> 3;
        const int cc = (t & 7) * 16;
        const float* srcp = H + (size_t)(row0 + r) * D + cc;
        #pragma unroll
        for (int j = 0; j < 16; j += 4) {
            float4 v = *(const float4*)(srcp + j);
            As[r * 132 + cc + j + 0] = v.x;
            As[r * 132 + cc + j + 1] = v.y;
            As[r * 132 + cc + j + 2] = v.z;
            As[r * 132 + cc + j + 3] = v.w;
        }
    }
    __syncthreads();

    const int wave = t >> 5;
    const int lane = t & 31;
    const float* W;
    float* Out;
    const float* bvec = nullptr;
    if (wave == 0)      { W = W1; Out = Hw1; }
    else if (wave == 1) { W = W2; Out = Hw2; }
    else if (wave == 2) { W = Wv; Out = Vv;  }
    else                { W = Wr; Out = R; bvec = b_res; }

    const int nlo   = lane & 15;         // A row (M) and B/C col (N)
    const int khalf = (lane >> 4) * 2;   // K sub-offset per half-wave

    for (int nt = 0; nt < 8; ++nt) {
        const int n0 = nt * 16;
        const float bv = bvec ? bvec[n0 + nlo] : 0.0f;
        v8f acc;
        #pragma unroll
        for (int i = 0; i < 8; ++i) acc[i] = bv;

        #pragma unroll 4
        for (int k0 = 0; k0 < D; k0 += 4) {
            const int ka = k0 + khalf;
            v2f a, b;
            a.x = As[nlo * 132 + ka];
            a.y = As[nlo * 132 + ka + 1];
            b.x = W[(size_t)ka * D + n0 + nlo];
            b.y = W[(size_t)(ka + 1) * D + n0 + nlo];
            acc = __builtin_amdgcn_wmma_f32_16x16x4_f32(
                false, a, false, b, (short)0, acc, false, false);
        }
        const int orow = row0 + (lane >> 4) * 8;
        const int ocol = n0 + nlo;
        #pragma unroll
        for (int i = 0; i < 8; ++i)
            Out[(size_t)(orow + i) * D + ocol] = acc[i];
    }
}

// ---------------------------------------------------------------------------
// 2) per-edge logits + LeakyReLU + scatter-max:  one wave per edge
// ---------------------------------------------------------------------------
__global__ __launch_bounds__(256) void edge_logits(
    const float* __restrict__ Hw1, const float* __restrict__ Hw2,
    const long long* __restrict__ ei, const float* __restrict__ P,
    const float* __restrict__ det, const float* __restrict__ W3,
    const float* __restrict__ W4, float* __restrict__ logits,
    float* __restrict__ mbuf, int E)
{
    __shared__ float W4s[NH * D];
    for (int i = threadIdx.x; i < NH * D; i += 256) W4s[i] = W4[i];
    __syncthreads();

    const int lane = threadIdx.x & 31;
    const int c0 = lane * 4;
    float w3v[4];
    #pragma unroll
    for (int j = 0; j < 4; ++j) w3v[j] = W3[c0 + j];

    int wid = blockIdx.x * 8 + (threadIdx.x >> 5);
    const int nw = gridDim.x * 8;
    for (int e = wid; e < E; e += nw) {
        const long long sn = ei[e];
        const long long dn = ei[(size_t)E + e];
        const float p = P[e];
        float4 h1 = *(const float4*)(Hw1 + (size_t)dn * D + c0);
        float4 h2 = *(const float4*)(Hw2 + (size_t)sn * D + c0);
        float eb[4];
        eb[0] = h1.x + h2.x + p * w3v[0];
        eb[1] = h1.y + h2.y + p * w3v[1];
        eb[2] = h1.z + h2.z + p * w3v[2];
        eb[3] = h1.w + h2.w + p * w3v[3];

        float lg[8];
        #pragma unroll
        for (int h = 0; h < 8; ++h) {
            const float* w = W4s + h * D + c0;
            lg[h] = eb[0] * w[0] + eb[1] * w[1] + eb[2] * w[2] + eb[3] * w[3];
        }
        #pragma unroll
        for (int h = 0; h < 8; ++h) {
            #pragma unroll
            for (int off = 16; off > 0; off >>= 1)
                lg[h] += __shfl_xor(lg[h], off, 32);
        }
        if (lane < 8) {
            float v = lg[0];
            if (lane == 1) v = lg[1];
            if (lane == 2) v = lg[2];
            if (lane == 3) v = lg[3];
            if (lane == 4) v = lg[4];
            if (lane == 5) v = lg[5];
            if (lane == 6) v = lg[6];
            if (lane == 7) v = lg[7];
            v += det[e];
            v = (v >= 0.0f) ? v : 0.2f * v;      // LeakyReLU(0.2)
            logits[(size_t)e * NH + lane] = v;
            atomicMaxF(mbuf + (size_t)dn * NH + lane, v);
        }
    }
}

// ---------------------------------------------------------------------------
// 3) exp(l - m[dst]) and per-(dst,head) sum
// ---------------------------------------------------------------------------
__global__ void edge_softmax_sum(const long long* __restrict__ ei,
                                 float* __restrict__ lgbuf,
                                 const float* __restrict__ mbuf,
                                 float* __restrict__ sbuf, int E)
{
    size_t idx = (size_t)blockIdx.x * blockDim.x + threadIdx.x;
    if (idx >= (size_t)E * NH) return;
    const size_t e = idx >> 3;
    const int h = (int)(idx & 7);
    const long long dn = ei[(size_t)E + e];
    const float ex = __expf(lgbuf[idx] - mbuf[(size_t)dn * NH + h]);
    lgbuf[idx] = ex;                         // reuse buffer for exp_e
    atomicAdd(sbuf + (size_t)dn * NH + h, ex);
}

// ---------------------------------------------------------------------------
// 4) weighted message scatter: one thread per (edge, channel)
// ---------------------------------------------------------------------------
__global__ void edge_scatter(const long long* __restrict__ ei,
                             const float* __restrict__ exbuf,
                             const float* __restrict__ sbuf,
                             const float* __restrict__ V,
                             float* __restrict__ agg, int E)
{
    size_t idx = (size_t)blockIdx.x * blockDim.x + threadIdx.x;
    if (idx >= (size_t)E * D) return;
    const size_t e = idx >> 7;
    const int c = (int)(idx & 127);
    const int h = c >> 4;
    const long long sn = ei[e];
    const long long dn = ei[(size_t)E + e];
    const float alpha = exbuf[e * NH + h] / (sbuf[(size_t)dn * NH + h] + 1e-12f);
    atomicAdd(agg + (size_t)dn * D + c, alpha * V[(size_t)sn * D + c]);
}

// ---------------------------------------------------------------------------
// 5) out = agg@Wout + b_out + R, then LayerNorm -> d_out
// ---------------------------------------------------------------------------
__global__ __launch_bounds__(128) void out_gemm_ln(
    const float* __restrict__ agg, const float* __restrict__ Wout,
    const float* __restrict__ b_out, const float* __restrict__ R,
    const float* __restrict__ gamma, const float* __restrict__ beta,
    float* __restrict__ out, int N)
{
    __shared__ float As[16 * 132];
    __shared__ float Xs[16 * 132];
    __shared__ float mu[16], rstd[16];
    const int t = threadIdx.x;
    const int row0 = blockIdx.x * 16;
    {
        const int r  = t >> 3;
        const int cc = (t & 7) * 16;
        const float* srcp = agg + (size_t)(row0 + r) * D + cc;
        #pragma unroll
        for (int j = 0; j < 16; j += 4) {
            float4 v = *(const float4*)(srcp + j);
            As[r * 132 + cc + j + 0] = v.x;
            As[r * 132 + cc + j + 1] = v.y;
            As[r * 132 + cc + j + 2] = v.z;
            As[r * 132 + cc + j + 3] = v.w;
        }
    }
    __syncthreads();

    const int wave = t >> 5;
    const int lane = t & 31;
    const int nlo   = lane & 15;
    const int khalf = (lane >> 4) * 2;

    for (int w = 0; w < 2; ++w) {
        const int n0 = (wave * 2 + w) * 16;
        const float bv = b_out[n0 + nlo];
        v8f acc;
        #pragma unroll
        for (int i = 0; i < 8; ++i) acc[i] = bv;

        #pragma unroll 4
        for (int k0 = 0; k0 < D; k0 += 4) {
            const int ka = k0 + khalf;
            v2f a, b;
            a.x = As[nlo * 132 + ka];
            a.y = As[nlo * 132 + ka + 1];
            b.x = Wout[(size_t)ka * D + n0 + nlo];
            b.y = Wout[(size_t)(ka + 1) * D + n0 + nlo];
            acc = __builtin_amdgcn_wmma_f32_16x16x4_f32(
                false, a, false, b, (short)0, acc, false, false);
        }
        const int lrow = (lane >> 4) * 8;
        const int ocol = n0 + nlo;
        #pragma unroll
        for (int i = 0; i < 8; ++i)
            Xs[(lrow + i) * 132 + ocol] =
                acc[i] + R[(size_t)(row0 + lrow + i) * D + ocol];
    }
    __syncthreads();

    if (t < 16) {
        float sum = 0.0f;
        for (int c = 0; c < D; ++c) sum += Xs[t * 132 + c];
        const float mean = sum * (1.0f / 128.0f);
        float var = 0.0f;
        for (int c = 0; c < D; ++c) {
            const float dlt = Xs[t * 132 + c] - mean;
            var += dlt * dlt;
        }
        var *= (1.0f / 128.0f);
        mu[t] = mean;
        rstd[t] = rsqrtf(var + 1e-5f);
    }
    __syncthreads();
    {
        const int r  = t >> 3;
        const int cc = (t & 7) * 16;
        #pragma unroll
        for (int j = 0; j < 16; ++j) {
            const int c = cc + j;
            out[(size_t)(row0 + r) * D + c] =
                (Xs[r * 132 + c] - mu[r]) * rstd[r] * gamma[c] + beta[c];
        }
    }
}

// ---------------------------------------------------------------------------
extern "C" void kernel_launch(void* const* d_in, const int* in_sizes, int n_in,
                              void* d_out, int out_size, void* d_ws, size_t ws_size,
                              hipStream_t stream) {
    const float*     H     = (const float*)d_in[0];
    const long long* ei    = (const long long*)d_in[1];   // jnp.int64
    const float*     P     = (const float*)d_in[2];
    const float*     det   = (const float*)d_in[3];
    const float*     W1    = (const float*)d_in[4];
    const float*     W2    = (const float*)d_in[5];
    const float*     W3    = (const float*)d_in[6];
    const float*     W4    = (const float*)d_in[7];
    const float*     Wv    = (const float*)d_in[8];
    const float*     Wout  = (const float*)d_in[9];
    const float*     b_out = (const float*)d_in[10];
    const float*     Wres  = (const float*)d_in[11];
    const float*     b_res = (const float*)d_in[12];
    const float*     gamma = (const float*)d_in[13];
    const float*     beta  = (const float*)d_in[14];

    const int N = in_sizes[0] / D;   // 50000 (divisible by 16)
    const int E = in_sizes[2];       // 800000

    float* ws  = (float*)d_ws;
    float* Hw1 = ws;
    float* Hw2 = Hw1 + (size_t)N * D;
    float* Vv  = Hw2 + (size_t)N * D;
    float* R   = Vv  + (size_t)N * D;
    float* agg = R   + (size_t)N * D;
    float* lgb = agg + (size_t)N * D;            // E*NH: logits, then exp_e
    float* mb  = lgb + (size_t)E * NH;           // N*NH segment max
    float* sb  = mb  + (size_t)N * NH;           // N*NH segment sum

    {
        size_t total = (size_t)N * D;
        init_ws<<<(int)((total + 255) / 256), 256, 0, stream>>>(agg, sb, mb, N);
    }
    node_gemm<<<N / 16, 128, 0, stream>>>(H, W1, W2, Wv, Wres, b_res,
                                          Hw1, Hw2, Vv, R, N);
    edge_logits<<<2048, 256, 0, stream>>>(Hw1, Hw2, ei, P, det, W3, W4,
                                          lgb, mb, E);
    {
        size_t tot = (size_t)E * NH;
        edge_softmax_sum<<<(int)((tot + 255) / 256), 256, 0, stream>>>(ei, lgb, mb, sb, E);
    }
    {
        size_t tot = (size_t)E * D;
        edge_scatter<<<(int)((tot + 255) / 256), 256, 0, stream>>>(ei, lgb, sb, Vv, agg, E);
    }
    out_gemm_ln<<<N / 16, 128, 0, stream>>>(agg, Wout, b_out, R, gamma, beta,
                                            (float*)d_out, N);
}